// TimingPropagation_84344567758887
// MI455X (gfx1250) — compile-verified
//
#include <hip/hip_runtime.h>

#define EPSF 1e-12f

// Dynamic index into 8 register-resident floats via cndmask chain (no scratch).
__device__ __forceinline__ float sel8(const float t[8], int i) {
    float r = t[0];
#pragma unroll
    for (int j = 1; j < 8; ++j) r = (i == j) ? t[j] : r;
    return r;
}

// gfx1250 hardware IEEE maximumNumber atomic, device scope, non-returning.
__device__ __forceinline__ void atomic_max_f32_dev(float* p, float v) {
    asm volatile("global_atomic_max_num_f32 %0, %1, off scope:SCOPE_DEV"
                 :
                 : "v"(p), "v"(v)
                 : "memory");
}

// Faithful port of _lut_entry for one arc.
__device__ __forceinline__ float lut_bilinear(
    float x_in, float y_in, int la,
    const float* __restrict__ values,  // [L, 64]
    const float* __restrict__ ttab,    // [L, 8] sorted ascending
    const float* __restrict__ ctab,    // [L, 8] sorted ascending
    const int*   __restrict__ dims)    // [L, 2]
{
    // 32B-aligned table rows -> two float4 loads each
    const float4* t4 = (const float4*)ttab;
    const float4* c4 = (const float4*)ctab;
    float4 ta = t4[2 * la + 0], tb = t4[2 * la + 1];
    float4 ca = c4[2 * la + 0], cb = c4[2 * la + 1];
    int2 d = ((const int2*)dims)[la];
    int tdim = d.x, cdim = d.y;

    float tt[8] = {ta.x, ta.y, ta.z, ta.w, tb.x, tb.y, tb.z, tb.w};
    float ct[8] = {ca.x, ca.y, ca.z, ca.w, cb.x, cb.y, cb.z, cb.w};

    // searchsorted(side='right') over the full 8-entry row = count of entries <= x
    int ti = 0, ci = 0;
#pragma unroll
    for (int j = 0; j < 8; ++j) {
        ti += (tt[j] <= x_in) ? 1 : 0;
        ci += (ct[j] <= y_in) ? 1 : 0;
    }
    int tmax = max(tdim - 1, 0);
    int cmax = max(cdim - 1, 0);
    ti = min(max(ti, 1), tmax);
    ci = min(max(ci, 1), cmax);
    int tlo = max(ti - 1, 0);
    int clo = max(ci - 1, 0);

    float t0 = sel8(tt, tlo), t1 = sel8(tt, ti);
    float c0 = sel8(ct, clo), c1 = sel8(ct, ci);

    // Corner gather: row stride is the ACTUAL cdim within the 64-entry row.
    const float* vrow = values + (size_t)la * 64;
    float v00 = vrow[tlo * cdim + clo];
    float v01 = vrow[tlo * cdim + ci];
    float v10 = vrow[ti * cdim + clo];
    float v11 = vrow[ti * cdim + ci];

    float t_int = t1 - t0;
    float c_int = c1 - c0;
    bool t_deg = fabsf(t_int) < EPSF;
    bool c_deg = fabsf(c_int) < EPSF;

    float x = fminf(fmaxf(x_in, t0), t1);
    float y = fminf(fmaxf(y_in, c0), c1);

    float wa = (t1 - x) * (c1 - y);
    float wb = (t1 - x) * (y - c0);
    float wc = (x - t0) * (c1 - y);
    float wd = (x - t0) * (y - c0);

    float t_safe = t_deg ? EPSF : t_int;
    float c_safe = c_deg ? EPSF : c_int;
    float denom = t_safe * c_safe;

    float lerp_c = fminf(fmaxf((y - c0) / fmaxf(c_safe, EPSF), 0.0f), 1.0f);
    float lerp_t = fminf(fmaxf((x - t0) / fmaxf(t_safe, EPSF), 0.0f), 1.0f);
    float val_t_deg = v00 + lerp_c * (v01 - v00);
    float val_c_deg = v00 + lerp_t * (v10 - v00);

    bool valid_denom = fabsf(denom) >= EPSF;
    float num = v00 * wa + v01 * wb + v10 * wc + v11 * wd;
    float raw = num / (valid_denom ? denom : 1.0f);
    float bilinear = (valid_denom && __builtin_isfinite(raw)) ? raw : v00;

    float fin = (t_deg && c_deg) ? v00
              : (t_deg ? val_t_deg
              : (c_deg ? val_c_deg : bilinear));

    bool valid = (tdim > 0) && (cdim > 0);
    return (valid && __builtin_isfinite(fin)) ? fin : 0.0f;
}

// d_out[0:P]=rAAT, [P:2P]=fAAT, [2P:3P]=rtran, [3P:4P]=ftran (max-accumulation seeds)
__global__ __launch_bounds__(256) void tp_init_kernel(
    const float* __restrict__ pin_rAAT, const float* __restrict__ pin_fAAT,
    const float* __restrict__ pin_rtran, const float* __restrict__ pin_ftran,
    float* __restrict__ out, int P)
{
    int i = blockIdx.x * blockDim.x + threadIdx.x;
    if (i >= P) return;
    size_t p = (size_t)P;
    out[i]         = pin_rAAT[i];
    out[i + p]     = pin_fAAT[i];
    out[i + 2 * p] = pin_rtran[i];
    out[i + 3 * p] = pin_ftran[i];
}

__global__ __launch_bounds__(256) void tp_arc_kernel(
    const float* __restrict__ pin_rAAT, const float* __restrict__ pin_fAAT,
    const float* __restrict__ pin_rtran, const float* __restrict__ pin_ftran,
    const float* __restrict__ pin_net_cap,
    const float* __restrict__ rd_values, const float* __restrict__ rd_trans,
    const float* __restrict__ rd_cap,    const int*   __restrict__ rd_dims,
    const float* __restrict__ fd_values, const float* __restrict__ fd_trans,
    const float* __restrict__ fd_cap,    const int*   __restrict__ fd_dims,
    const float* __restrict__ rt_values, const float* __restrict__ rt_trans,
    const float* __restrict__ rt_cap,    const int*   __restrict__ rt_dims,
    const float* __restrict__ ft_values, const float* __restrict__ ft_trans,
    const float* __restrict__ ft_cap,    const int*   __restrict__ ft_dims,
    const int* __restrict__ arc_in, const int* __restrict__ arc_out,
    const int* __restrict__ lib_idx,
    float* __restrict__ out, int P, int A)
{
    int i = blockIdx.x * blockDim.x + threadIdx.x;
    if (i >= A) return;

    // Streamed once -> non-temporal so they don't evict hot LUT/pin data.
    int ip = __builtin_nontemporal_load(arc_in + i);
    int op = __builtin_nontemporal_load(arc_out + i);
    int la = __builtin_nontemporal_load(lib_idx + i);

    // Gathers (hot in L2: all pin arrays total 40MB << 192MB).
    float it_r  = pin_rtran[ip];
    float it_f  = pin_ftran[ip];
    float oc    = pin_net_cap[op];
    float aat_r = pin_rAAT[ip];
    float aat_f = pin_fAAT[ip];

    float r_del = lut_bilinear(it_r, oc, la, rd_values, rd_trans, rd_cap, rd_dims);
    float f_del = lut_bilinear(it_f, oc, la, fd_values, fd_trans, fd_cap, fd_dims);
    float r_trn = lut_bilinear(it_r, oc, la, rt_values, rt_trans, rt_cap, rt_dims);
    float f_trn = lut_bilinear(it_f, oc, la, ft_values, ft_trans, ft_cap, ft_dims);

    size_t p = (size_t)P;
    atomic_max_f32_dev(out + op,         aat_r + r_del);
    atomic_max_f32_dev(out + op + p,     aat_f + f_del);
    atomic_max_f32_dev(out + op + 2 * p, r_trn);
    atomic_max_f32_dev(out + op + 3 * p, f_trn);
}

extern "C" void kernel_launch(void* const* d_in, const int* in_sizes, int n_in,
                              void* d_out, int out_size, void* d_ws, size_t ws_size,
                              hipStream_t stream) {
    const float* pin_rAAT    = (const float*)d_in[0];
    const float* pin_fAAT    = (const float*)d_in[1];
    const float* pin_rtran   = (const float*)d_in[2];
    const float* pin_ftran   = (const float*)d_in[3];
    const float* pin_net_cap = (const float*)d_in[4];

    const float* rd_values = (const float*)d_in[5];
    const float* rd_trans  = (const float*)d_in[6];
    const float* rd_cap    = (const float*)d_in[7];
    const float* fd_values = (const float*)d_in[8];
    const float* fd_trans  = (const float*)d_in[9];
    const float* fd_cap    = (const float*)d_in[10];
    const float* rt_values = (const float*)d_in[11];
    const float* rt_trans  = (const float*)d_in[12];
    const float* rt_cap    = (const float*)d_in[13];
    const float* ft_values = (const float*)d_in[14];
    const float* ft_trans  = (const float*)d_in[15];
    const float* ft_cap    = (const float*)d_in[16];

    const int* rd_dims = (const int*)d_in[17];
    const int* fd_dims = (const int*)d_in[18];
    const int* rt_dims = (const int*)d_in[19];
    const int* ft_dims = (const int*)d_in[20];

    const int* arc_in  = (const int*)d_in[21];
    const int* arc_out = (const int*)d_in[22];
    const int* lib_idx = (const int*)d_in[23];

    int P = in_sizes[0];
    int A = in_sizes[21];
    float* out = (float*)d_out;

    const int TB = 256;  // 8 wave32 waves per block
    tp_init_kernel<<<(P + TB - 1) / TB, TB, 0, stream>>>(
        pin_rAAT, pin_fAAT, pin_rtran, pin_ftran, out, P);

    tp_arc_kernel<<<(A + TB - 1) / TB, TB, 0, stream>>>(
        pin_rAAT, pin_fAAT, pin_rtran, pin_ftran, pin_net_cap,
        rd_values, rd_trans, rd_cap, rd_dims,
        fd_values, fd_trans, fd_cap, fd_dims,
        rt_values, rt_trans, rt_cap, rt_dims,
        ft_values, ft_trans, ft_cap, ft_dims,
        arc_in, arc_out, lib_idx,
        out, P, A);
}